// SplineConv_56908316672604
// MI455X (gfx1250) — compile-verified
//
#include <hip/hip_runtime.h>

typedef __attribute__((ext_vector_type(2))) float v2f;
typedef __attribute__((ext_vector_type(8))) float v8f;

#define D_FEAT 128
#define TN 8            // D_FEAT / 16

// ---------------------------------------------------------------------------
// Kernel 1: fused node-level GEMMs via V_WMMA_F32_16X16X4_F32.
//   h0 = x @ W[0]        (to ws)
//   h1 = x @ W[1]        (to ws)
//   out = x @ W_root + b (r term, straight into d_out)
// One wave = one 16x16 tile of all three outputs (shared A fragment).
// A-frag (16x4 f32): lanes 0-15 -> M=0..15, VGPR0/1 = K{0,1}; lanes 16-31 = K{2,3}.
// B-frag (4x16 f32): mirrored, N across lanes.
// C/D (16x16 f32): VGPR v, lane l -> M = v + 8*(l>>4), N = l&15.
// ---------------------------------------------------------------------------
__global__ void __launch_bounds__(256)
spline_gemm_wmma(const float* __restrict__ x, const float* __restrict__ W,
                 const float* __restrict__ Wr, const float* __restrict__ bias,
                 float* __restrict__ h0, float* __restrict__ h1,
                 float* __restrict__ out, int nTiles) {
  const int lane = threadIdx.x & 31;
  const int tile = blockIdx.x * 8 + (threadIdx.x >> 5);
  if (tile >= nTiles) return;
  const int tm = tile >> 3;          // tile / TN
  const int tn = tile & (TN - 1);
  const int row0 = tm * 16;
  const int col0 = tn * 16;
  const int m  = lane & 15;          // M index (A) == N index (B)
  const int kh = lane >> 4;          // K-half selector

  const float* W0 = W;
  const float* W1 = W + D_FEAT * D_FEAT;

  v8f c0 = {}, c1 = {}, cr = {};
  for (int kk = 0; kk < D_FEAT; kk += 4) {
    const int ka = kk + 2 * kh;
    // A fragment: two consecutive K elements of row (row0+m)
    v2f a = *(const v2f*)(x + (size_t)(row0 + m) * D_FEAT + ka);
    // B fragments: rows ka, ka+1 of each weight matrix, column col0+m
    const float* w0p = W0 + (size_t)ka * D_FEAT + col0 + m;
    const float* w1p = W1 + (size_t)ka * D_FEAT + col0 + m;
    const float* wrp = Wr + (size_t)ka * D_FEAT + col0 + m;
    v2f b0; b0.x = w0p[0]; b0.y = w0p[D_FEAT];
    v2f b1; b1.x = w1p[0]; b1.y = w1p[D_FEAT];
    v2f br; br.x = wrp[0]; br.y = wrp[D_FEAT];

    c0 = __builtin_amdgcn_wmma_f32_16x16x4_f32(false, a, false, b0, (short)0, c0, false, false);
    c1 = __builtin_amdgcn_wmma_f32_16x16x4_f32(false, a, false, b1, (short)0, c1, false, false);
    cr = __builtin_amdgcn_wmma_f32_16x16x4_f32(false, a, false, br, (short)0, cr, false, false);
  }

  const float bb = bias[col0 + m];
#pragma unroll
  for (int v = 0; v < 8; ++v) {
    const size_t rr = (size_t)(row0 + v + 8 * kh) * D_FEAT + (col0 + m);
    h0[rr]  = c0[v];
    h1[rr]  = c1[v];
    out[rr] = cr[v] + bb;   // root term + bias, aggregation added later
  }
}

// ---------------------------------------------------------------------------
// Kernel 0: zero agg + deg (contiguous region of ws)
// ---------------------------------------------------------------------------
__global__ void spline_zero(float* __restrict__ p, size_t n) {
  size_t i = (size_t)blockIdx.x * blockDim.x + threadIdx.x;
  const size_t stride = (size_t)gridDim.x * blockDim.x;
  for (; i < n; i += stride) p[i] = 0.0f;
}

// ---------------------------------------------------------------------------
// Kernel 2: per-edge blend + scatter. One wave per edge; each lane owns a
// float4 slice of the 128-wide feature row. h0/h1 (51.2 MB) live in the
// 192 MB L2, so the random gathers are L2-resident. fp32 HW atomics for the
// segment sum; lane 0 accumulates degree.
// ---------------------------------------------------------------------------
__global__ void __launch_bounds__(256)
spline_edge(const float* __restrict__ h0, const float* __restrict__ h1,
            const int* __restrict__ src, const int* __restrict__ dst,
            const float* __restrict__ u, float* __restrict__ agg,
            float* __restrict__ deg, int E) {
  const int wave = (int)(((size_t)blockIdx.x * blockDim.x + threadIdx.x) >> 5);
  const int lane = threadIdx.x & 31;
  if (wave >= E) return;

  const int s = src[wave];
  const int d = dst[wave];
  float w1 = u[wave];
  w1 = fminf(fmaxf(w1, 0.0f), 1.0f);
  const float w0 = 1.0f - w1;

  const float4 a = *((const float4*)(h0 + (size_t)s * D_FEAT) + lane);
  const float4 b = *((const float4*)(h1 + (size_t)s * D_FEAT) + lane);

  float4 msg;
  msg.x = w0 * a.x + w1 * b.x;
  msg.y = w0 * a.y + w1 * b.y;
  msg.z = w0 * a.z + w1 * b.z;
  msg.w = w0 * a.w + w1 * b.w;

  float* o = agg + (size_t)d * D_FEAT + lane * 4;
  unsafeAtomicAdd(o + 0, msg.x);
  unsafeAtomicAdd(o + 1, msg.y);
  unsafeAtomicAdd(o + 2, msg.z);
  unsafeAtomicAdd(o + 3, msg.w);
  if (lane == 0) unsafeAtomicAdd(deg + d, 1.0f);
}

// ---------------------------------------------------------------------------
// Kernel 3: out += agg / max(deg, 1)
// ---------------------------------------------------------------------------
__global__ void spline_final(float* __restrict__ out, const float* __restrict__ agg,
                             const float* __restrict__ deg, int total) {
  const int i = (int)((size_t)blockIdx.x * blockDim.x + threadIdx.x);
  if (i < total) {
    const float dv = fmaxf(deg[i >> 7], 1.0f);   // 128 = 1<<7 features
    out[i] += agg[i] / dv;
  }
}

extern "C" void kernel_launch(void* const* d_in, const int* in_sizes, int n_in,
                              void* d_out, int out_size, void* d_ws, size_t ws_size,
                              hipStream_t stream) {
  const float* x    = (const float*)d_in[0];
  const int*   ei   = (const int*)  d_in[1];
  const float* u    = (const float*)d_in[2];
  const float* W    = (const float*)d_in[3];
  const float* Wr   = (const float*)d_in[4];
  const float* bias = (const float*)d_in[5];
  float* out = (float*)d_out;

  const int N = in_sizes[0] / D_FEAT;   // 50000
  const int E = in_sizes[2];            // 1600000 (edge_attr element count)
  const int* src = ei;
  const int* dst = ei + E;

  // workspace layout: h0 | h1 | agg | deg
  float* h0  = (float*)d_ws;
  float* h1  = h0  + (size_t)N * D_FEAT;
  float* agg = h1  + (size_t)N * D_FEAT;
  float* deg = agg + (size_t)N * D_FEAT;

  // zero agg + deg (contiguous)
  spline_zero<<<2048, 256, 0, stream>>>(agg, (size_t)N * D_FEAT + (size_t)N);

  // fused GEMMs (WMMA)
  const int nTiles = ((N + 15) / 16) * TN;          // 25000
  spline_gemm_wmma<<<(nTiles + 7) / 8, 256, 0, stream>>>(x, W, Wr, bias, h0, h1, out, nTiles);

  // edge gather/blend/scatter: one wave per edge
  {
    const long long threads = (long long)E * 32;
    const int blocks = (int)((threads + 255) / 256);
    spline_edge<<<blocks, 256, 0, stream>>>(h0, h1, src, dst, u, agg, deg, E);
  }

  // finalize: mean aggregation + root term already in out
  const int total = N * D_FEAT;
  spline_final<<<(total + 255) / 256, 256, 0, stream>>>(out, agg, deg, total);
}